// GeneNetworkGreensFunction_7378753815356
// MI455X (gfx1250) — compile-verified
//
#include <hip/hip_runtime.h>
#include <math.h>

#define N_GENES   4096
#define RANK      32
#define K_PACK    64          // [re | im] packed K dimension
#define CLAMP_MAX 10.0f

typedef float v2f __attribute__((ext_vector_type(2)));
typedef float v8f __attribute__((ext_vector_type(8)));

// ---------------------------------------------------------------------------
// Kernel 1: per-row resolvent diagonal inverse + V = A^-1 U, packed directly
// into the GEMM B operands:  Bre[i] = [Vr | -Vi],  Bim[i] = [Vi | Vr].
// Also zeroes the 32x32 complex accumulator S.
// ---------------------------------------------------------------------------
__global__ __launch_bounds__(256) void prep_kernel(
    const float* __restrict__ omega_p, const float* __restrict__ U,
    const float* __restrict__ Hdiag,   const float* __restrict__ logeta_p,
    float* __restrict__ invAre, float* __restrict__ invAim,
    float* __restrict__ Bre,    float* __restrict__ Bim,
    float* __restrict__ S)
{
    int i = blockIdx.x * 256 + threadIdx.x;            // 0..4095
    float omega = omega_p[0];
    float eta   = expf(logeta_p[0]);
    float ar    = omega - Hdiag[i];                    // Re(z - d_i)
    float den   = ar * ar + eta * eta;
    float iar   =  ar  / den;                          // Re(1/a_i)
    float iai   = -eta / den;                          // Im(1/a_i)
    invAre[i] = iar;
    invAim[i] = iai;

    const float* urow = U + (size_t)i * RANK;
    float* bre = Bre + (size_t)i * K_PACK;
    float* bim = Bim + (size_t)i * K_PACK;
#pragma unroll
    for (int k = 0; k < RANK; ++k) {
        float u  = urow[k];
        float vr = iar * u;
        float vi = iai * u;
        bre[k]        =  vr;
        bre[RANK + k] = -vi;
        bim[k]        =  vi;
        bim[RANK + k] =  vr;
    }
    if (blockIdx.x == 0) {
        for (int e = threadIdx.x; e < 2 * RANK * RANK; e += 256) S[e] = 0.0f;
    }
}

// ---------------------------------------------------------------------------
// Kernel 2: S_{kl} = sum_i U_{ik} * V_{il}   (complex; V read from Bre/Bim)
// 32 blocks x 128 rows each, float atomics into S.
// ---------------------------------------------------------------------------
__global__ __launch_bounds__(256) void s_partial_kernel(
    const float* __restrict__ U, const float* __restrict__ Bre,
    const float* __restrict__ Bim, float* __restrict__ S)
{
    int i0 = blockIdx.x * 128;
    int t  = threadIdx.x;
#pragma unroll
    for (int q = 0; q < 4; ++q) {
        int p = t + q * 256;           // pair index 0..1023
        int k = p & 31;
        int l = p >> 5;
        float sr = 0.0f, si = 0.0f;
        for (int i = i0; i < i0 + 128; ++i) {
            float u  = U  [(size_t)i * RANK   + k];
            float vr = Bre[(size_t)i * K_PACK + l];   // V_re
            float vi = Bim[(size_t)i * K_PACK + l];   // V_im
            sr += u * vr;
            si += u * vi;
        }
        atomicAdd(&S[2 * p + 0], sr);
        atomicAdd(&S[2 * p + 1], si);
    }
}

// ---------------------------------------------------------------------------
// Kernel 3: W = (I - S)^-1, 32x32 complex Gauss-Jordan w/ partial pivoting.
// One workgroup; augmented [M | I] lives in LDS (32x64 float2 = 16 KB).
// ---------------------------------------------------------------------------
__global__ __launch_bounds__(1024) void invert32_kernel(
    const float* __restrict__ S, float* __restrict__ W)
{
    __shared__ float2 Aug[32][64];
    __shared__ float2 fac[32];
    __shared__ int   pivRow;
    __shared__ float2 pivInv;
    int t = threadIdx.x;

    for (int e = t; e < 32 * 64; e += 1024) {
        int r = e >> 6, c = e & 63;
        float2 v;
        if (c < 32) {
            v.x = (r == c ? 1.0f : 0.0f) - S[2 * (r * 32 + c) + 0];
            v.y =                        - S[2 * (r * 32 + c) + 1];
        } else {
            v.x = (c - 32 == r) ? 1.0f : 0.0f;
            v.y = 0.0f;
        }
        Aug[r][c] = v;
    }
    __syncthreads();

    for (int p = 0; p < 32; ++p) {
        if (t == 0) {
            int best = p; float bm = -1.0f;
            for (int r = p; r < 32; ++r) {
                float2 v = Aug[r][p];
                float  m = v.x * v.x + v.y * v.y;
                if (m > bm) { bm = m; best = r; }
            }
            pivRow = best;
        }
        __syncthreads();
        int pr = pivRow;
        if (pr != p && t < 64) {
            float2 tmp  = Aug[p][t];
            Aug[p][t]   = Aug[pr][t];
            Aug[pr][t]  = tmp;
        }
        __syncthreads();
        if (t == 0) {
            float2 d = Aug[p][p];
            float den = d.x * d.x + d.y * d.y;
            pivInv.x =  d.x / den;
            pivInv.y = -d.y / den;
        }
        __syncthreads();
        if (t < 64) {
            float2 v = Aug[p][t];
            float2 w;
            w.x = v.x * pivInv.x - v.y * pivInv.y;
            w.y = v.x * pivInv.y + v.y * pivInv.x;
            Aug[p][t] = w;
        }
        __syncthreads();
        if (t < 32) fac[t] = Aug[t][p];
        __syncthreads();
        for (int e = t; e < 32 * 64; e += 1024) {
            int r = e >> 6, c = e & 63;
            if (r != p) {
                float2 f  = fac[r];
                float2 pv = Aug[p][c];
                float2 v  = Aug[r][c];
                v.x -= f.x * pv.x - f.y * pv.y;
                v.y -= f.x * pv.y + f.y * pv.x;
                Aug[r][c] = v;
            }
        }
        __syncthreads();
    }
    if (t < 32 * 32) {
        int r = t >> 5, c = t & 31;
        W[2 * t + 0] = Aug[r][32 + c].x;
        W[2 * t + 1] = Aug[r][32 + c].y;
    }
}

// ---------------------------------------------------------------------------
// Kernel 4: P = V * W (4096x32 complex), packed as Apack[i] = [Pr | Pi].
// ---------------------------------------------------------------------------
__global__ __launch_bounds__(256) void apack_kernel(
    const float* __restrict__ Bre, const float* __restrict__ Bim,
    const float* __restrict__ W,   float* __restrict__ Apack)
{
    __shared__ float Wsh[2 * RANK * RANK];
    for (int e = threadIdx.x; e < 2 * RANK * RANK; e += 256) Wsh[e] = W[e];
    __syncthreads();

    int gid = blockIdx.x * 256 + threadIdx.x;   // 0..131071
    int i = gid >> 5, k = gid & 31;
    const float* bre = Bre + (size_t)i * K_PACK;
    const float* bim = Bim + (size_t)i * K_PACK;
    float pr = 0.0f, pi = 0.0f;
#pragma unroll
    for (int l = 0; l < RANK; ++l) {
        float vr = bre[l];                      // V_re[i][l]
        float vi = bim[l];                      // V_im[i][l]
        float wr = Wsh[2 * (l * 32 + k) + 0];
        float wi = Wsh[2 * (l * 32 + k) + 1];
        pr += vr * wr - vi * wi;
        pi += vr * wi + vi * wr;
    }
    Apack[(size_t)i * K_PACK + k]        = pr;
    Apack[(size_t)i * K_PACK + RANK + k] = pi;
}

// ---------------------------------------------------------------------------
// Kernel 5: G = diag(invA) + P V^T via WMMA f32 16x16x4, then min(|G|, 10).
// One wave per 16x16 tile; two f32 accumulators (re/im); K = 64 packed.
// A-operand layout (ISA 7.12.2): lane L -> M = L&15, VGPRs hold K = 2*(L>>4),
// 2*(L>>4)+1. B mirrors with N = L&15; since we compute A*B^T with both
// operands row-major [4096 x 64], A and B per-lane loads are identical.
// ---------------------------------------------------------------------------
__global__ __launch_bounds__(256) void gemm_wmma_kernel(
    const float* __restrict__ Apack, const float* __restrict__ Bre,
    const float* __restrict__ Bim,   const float* __restrict__ invAre,
    const float* __restrict__ invAim, float* __restrict__ out)
{
    int wave = (blockIdx.x << 3) | (threadIdx.x >> 5);   // 0..65535
    int tr = wave >> 8;                                  // tile row 0..255
    int tc = wave & 255;                                 // tile col 0..255
    int row0 = tr << 4, col0 = tc << 4;

    int lane = threadIdx.x & 31;
    int half = lane >> 4;        // 0: K pair {0,1}; 1: K pair {2,3}
    int mn   = lane & 15;        // M for A, N for B

    const float* aRow   = Apack + (size_t)(row0 + mn) * K_PACK + 2 * half;
    const float* bRowRe = Bre   + (size_t)(col0 + mn) * K_PACK + 2 * half;
    const float* bRowIm = Bim   + (size_t)(col0 + mn) * K_PACK + 2 * half;

    v8f cre = {0.f,0.f,0.f,0.f,0.f,0.f,0.f,0.f};
    v8f cim = {0.f,0.f,0.f,0.f,0.f,0.f,0.f,0.f};
#pragma unroll
    for (int kk = 0; kk < K_PACK; kk += 4) {
        v2f a   = *(const v2f*)(aRow   + kk);
        v2f bre = *(const v2f*)(bRowRe + kk);
        v2f bim = *(const v2f*)(bRowIm + kk);
        cre = __builtin_amdgcn_wmma_f32_16x16x4_f32(
                  false, a, false, bre, (short)0, cre, false, false);
        cim = __builtin_amdgcn_wmma_f32_16x16x4_f32(
                  false, a, false, bim, (short)0, cim, false, false);
    }

    // C/D layout: VGPR v, lane L -> M = v + 8*(L>>4), N = L&15
#pragma unroll
    for (int v = 0; v < 8; ++v) {
        int m    = v + (half << 3);
        int grow = row0 + m;
        int gcol = col0 + mn;
        float re = cre[v];
        float im = cim[v];
        if (grow == gcol) {          // diagonal A^-1 term of Woodbury
            re += invAre[grow];
            im += invAim[grow];
        }
        float mag = sqrtf(re * re + im * im);
        out[(size_t)grow * N_GENES + gcol] = fminf(mag, CLAMP_MAX);
    }
}

// ---------------------------------------------------------------------------
// Host-side launcher (graph-capture safe: only kernel launches on `stream`).
// Workspace layout (floats): invAre[4096] invAim[4096] S[2048] W[2048]
//                            Apack[256K]  Bre[256K]    Bim[256K]  (~3.2 MB)
// ---------------------------------------------------------------------------
extern "C" void kernel_launch(void* const* d_in, const int* in_sizes, int n_in,
                              void* d_out, int out_size, void* d_ws, size_t ws_size,
                              hipStream_t stream)
{
    (void)in_sizes; (void)n_in; (void)out_size; (void)ws_size;

    const float* omega  = (const float*)d_in[0];
    const float* U      = (const float*)d_in[1];   // 4096 x 32
    const float* Hdiag  = (const float*)d_in[2];   // 4096
    const float* logeta = (const float*)d_in[3];
    float* out = (float*)d_out;

    float* ws     = (float*)d_ws;
    float* invAre = ws;                 // 4096
    float* invAim = ws + 4096;          // 4096
    float* S      = ws + 8192;          // 2048
    float* W      = ws + 10240;         // 2048
    float* Apack  = ws + 12288;         // 4096*64
    float* Bre    = ws + 12288 + 262144;
    float* Bim    = ws + 12288 + 2 * 262144;

    prep_kernel     <<<N_GENES / 256, 256, 0, stream>>>(omega, U, Hdiag, logeta,
                                                        invAre, invAim, Bre, Bim, S);
    s_partial_kernel<<<32,  256,  0, stream>>>(U, Bre, Bim, S);
    invert32_kernel <<<1,   1024, 0, stream>>>(S, W);
    apack_kernel    <<<512, 256,  0, stream>>>(Bre, Bim, W, Apack);
    gemm_wmma_kernel<<<(N_GENES / 16) * (N_GENES / 16) / 8, 256, 0, stream>>>(
        Apack, Bre, Bim, invAre, invAim, out);
}